// Pos_embedding_3461743640584
// MI455X (gfx1250) — compile-verified
//
#include <hip/hip_runtime.h>
#include <math.h>

// out[r,c] = x[r,c]*sqrt(512) + (c even ? sin : cos)( r * 10000^(-c/256) )
// 131072 x 512 f32, pure streaming: 512 MB traffic -> HBM-bound (~22us @ 23.3 TB/s).
// Strategy: float4 (b128) NT streaming loads/stores, wave32, hardware v_sin/v_cos
// with double-float range reduction in "revolutions" so accuracy holds at r=131071.

typedef float v4f __attribute__((ext_vector_type(4)));

#define DIM       512
#define SCALE_F   22.627416997969522f   /* sqrt(512) */

// ---------------------------------------------------------------------------
// Setup: build g(c) = 10000^(-c/256) / (2*pi) as a hi/lo float pair per column.
// 512 threads, runs once per launch (negligible), result lives in d_ws (4 KB).
// Layout: tab[0..511] = hi, tab[512..1023] = lo.
// ---------------------------------------------------------------------------
__global__ void pe_table_kernel(float* __restrict__ tab) {
    int c = threadIdx.x;                          // 0..511
    // ln(10000)/256 and 1/(2*pi), computed in f64 for a ~2^-46-accurate g(c)
    double gd = exp(-(double)c * (9.210340371976184 / 256.0))
              * 0.15915494309189535;
    float hi = (float)gd;
    float lo = (float)(gd - (double)hi);
    tab[c]       = hi;
    tab[DIM + c] = lo;
}

// ---------------------------------------------------------------------------
// Main kernel: one float4 per lane. gid = r*128 + c4, so the flat float4 index
// is gid itself. Columns 4*c4 .. 4*c4+3 alternate sin,cos,sin,cos.
// Range reduction: frac(r*g) via TwoProdFMA -> |f| <= ~0.5 revolutions, then
// hardware sin/cos (v_sin_f32/v_cos_f32 take input in revolutions).
// ---------------------------------------------------------------------------
__global__ __launch_bounds__(256) void pos_embed_kernel(
    const v4f* __restrict__ x,
    v4f* __restrict__ out,
    const v4f* __restrict__ tab_hi,
    const v4f* __restrict__ tab_lo)
{
    unsigned gid = blockIdx.x * 256u + threadIdx.x;
    unsigned r   = gid >> 7;      // row (128 float4 groups per 512-wide row)
    unsigned c4  = gid & 127u;    // float4 group within the row
    float rf = (float)r;          // exact: r < 2^17

    v4f xv = __builtin_nontemporal_load(&x[gid]);   // streamed once: NT
    v4f gh = tab_hi[c4];                            // 4 KB table: cache-resident
    v4f gl = tab_lo[c4];

    v4f o;

    // lane helper expanded manually (sin for even cols, cos for odd cols)
    {   // col 4*c4 + 0 : sin
        float p = rf * gh.x;
        float e = fmaf(rf, gh.x, -p);                       // exact residual
        float f = (p - rintf(p)) + fmaf(rf, gl.x, e);       // frac revolutions
        o.x = fmaf(xv.x, SCALE_F, __builtin_amdgcn_sinf(f));
    }
    {   // col 4*c4 + 1 : cos
        float p = rf * gh.y;
        float e = fmaf(rf, gh.y, -p);
        float f = (p - rintf(p)) + fmaf(rf, gl.y, e);
        o.y = fmaf(xv.y, SCALE_F, __builtin_amdgcn_cosf(f));
    }
    {   // col 4*c4 + 2 : sin
        float p = rf * gh.z;
        float e = fmaf(rf, gh.z, -p);
        float f = (p - rintf(p)) + fmaf(rf, gl.z, e);
        o.z = fmaf(xv.z, SCALE_F, __builtin_amdgcn_sinf(f));
    }
    {   // col 4*c4 + 3 : cos
        float p = rf * gh.w;
        float e = fmaf(rf, gh.w, -p);
        float f = (p - rintf(p)) + fmaf(rf, gl.w, e);
        o.w = fmaf(xv.w, SCALE_F, __builtin_amdgcn_cosf(f));
    }

    __builtin_nontemporal_store(o, &out[gid]);      // write-once: NT store
}

// ---------------------------------------------------------------------------
extern "C" void kernel_launch(void* const* d_in, const int* in_sizes, int n_in,
                              void* d_out, int out_size, void* d_ws, size_t ws_size,
                              hipStream_t stream)
{
    (void)n_in; (void)out_size; (void)ws_size;
    const float* x   = (const float*)d_in[0];
    float*       out = (float*)d_out;
    float*       tab = (float*)d_ws;                // needs 4 KB scratch

    // Build the per-column hi/lo divisor table (runs every call; deterministic).
    pe_table_kernel<<<1, DIM, 0, stream>>>(tab);

    // 131072*512 floats = 16,777,216 float4 groups = 65536 blocks of 256.
    int n4 = in_sizes[0] / 4;
    dim3 grid((unsigned)(n4 / 256));
    pos_embed_kernel<<<grid, 256, 0, stream>>>(
        (const v4f*)x, (v4f*)out,
        (const v4f*)tab, (const v4f*)(tab + DIM));
}